// SimCCHead_27530740367583
// MI455X (gfx1250) — compile-verified
//
#include <hip/hip_runtime.h>

#define IN_CH     512
#define OUT_CH    5376          // 21 * 256
#define BATCH     64
#define HW        4096          // 64*64
#define LDSPITCH  516           // 512 + 4 floats pad -> conflict-free row reads

typedef __attribute__((ext_vector_type(2))) float v2f;
typedef __attribute__((ext_vector_type(4))) float f4;
typedef __attribute__((ext_vector_type(8))) float v8f;

// ---------------------------------------------------------------------------
// Kernel 1: spatial mean.  One block per (b,c) pair; 256 threads reduce 4096
// floats via 4x b128 non-temporal loads each (streamed once, keep L2 clean).
// ---------------------------------------------------------------------------
__global__ void __launch_bounds__(256)
simcc_mean_kernel(const float* __restrict__ z, float* __restrict__ zm)
{
    const int bc = blockIdx.x;                       // b*512 + c
    const f4* p = (const f4*)(z + (size_t)bc * HW);  // 1024 float4s

    float s = 0.0f;
#pragma unroll
    for (int i = 0; i < 4; ++i) {
        f4 v = __builtin_nontemporal_load(&p[threadIdx.x + 256 * i]);
        s += (v.x + v.y) + (v.z + v.w);
    }

    // wave32 reduction
#pragma unroll
    for (int off = 16; off > 0; off >>= 1)
        s += __shfl_down(s, off, 32);

    __shared__ float red[8];
    if ((threadIdx.x & 31) == 0) red[threadIdx.x >> 5] = s;
    __syncthreads();

    if (threadIdx.x == 0) {
        float t = 0.0f;
#pragma unroll
        for (int i = 0; i < 8; ++i) t += red[i];
        zm[bc] = t * (1.0f / (float)HW);
    }
}

// ---------------------------------------------------------------------------
// Kernel 2: dual GEMM  out[mat][m][n] = sum_k zm[m][k] * w[mat][n][k] + b[n]
// using V_WMMA_F32_16X16X4_F32.  Block = 4 waves, one shared M-tile in LDS,
// each wave owns one 16x16 N-tile, K swept 512 in steps of 4 (128 WMMAs).
// ---------------------------------------------------------------------------
__global__ void __launch_bounds__(128)
simcc_gemm_kernel(const float* __restrict__ zm,
                  const float* __restrict__ wx, const float* __restrict__ bx,
                  const float* __restrict__ wy, const float* __restrict__ by,
                  float* __restrict__ out)
{
    __shared__ float smA[16 * LDSPITCH];

    // grid = 84 n-groups * 4 m-tiles * 2 matrices = 672 blocks
    const int g      = blockIdx.x;
    const int ngroup = g % 84;
    const int mtile  = (g / 84) & 3;
    const int mat    = g / (84 * 4);

    const float* __restrict__ w    = mat ? wy : wx;
    const float* __restrict__ bias = mat ? by : bx;

    const int mbase = mtile * 16;

    // Cooperative stage of zm[mbase..mbase+15][0..511] into padded LDS.
    for (int f = threadIdx.x; f < 16 * 128; f += 128) {   // f in float4 units
        const int row = f >> 7;
        const int c4  = f & 127;
        const f4 v = *((const f4*)(zm + (size_t)(mbase + row) * IN_CH) + c4);
        float* d = &smA[row * LDSPITCH + c4 * 4];
        d[0] = v.x; d[1] = v.y; d[2] = v.z; d[3] = v.w;
    }
    __syncthreads();

    const int wave = threadIdx.x >> 5;
    const int lane = threadIdx.x & 31;
    const int lrow = lane & 15;
    const int kofs = (lane >> 4) << 1;      // lanes 0-15: K+0/K+1, lanes 16-31: K+2/K+3
    const int nbase = (ngroup * 4 + wave) * 16;
    const int n = nbase + lrow;

    const float* __restrict__ wrow = w + (size_t)n * IN_CH + kofs;
    const float* __restrict__ arow = &smA[lrow * LDSPITCH + kofs];

    // C/D: value (m = mbase + r + (lane>=16 ? 8:0), n = nbase + lane%16) in VGPR r.
    // Bias depends only on n -> identical across all 8 accumulator registers.
    const float bv = bias[n];
    v8f acc = { bv, bv, bv, bv, bv, bv, bv, bv };

#pragma unroll 8
    for (int k = 0; k < IN_CH; k += 4) {
        v2f a; a.x = arow[k]; a.y = arow[k + 1];   // ds_load (A 16x4 fragment)
        v2f b; b.x = wrow[k]; b.y = wrow[k + 1];   // global_load (B 4x16 fragment)
        acc = __builtin_amdgcn_wmma_f32_16x16x4_f32(
            /*neg_a=*/false, a, /*neg_b=*/false, b,
            /*c_mod=*/(short)0, acc, /*reuse_a=*/false, /*reuse_b=*/false);
    }

    // Store into concatenated (x, y): out[mat*64*5376 + m*5376 + n]
    const int mofs = mbase + ((lane >> 4) ? 8 : 0);
    float* op = out + (size_t)mat * BATCH * OUT_CH
                    + (size_t)mofs * OUT_CH + nbase + lrow;
#pragma unroll
    for (int r = 0; r < 8; ++r)
        op[(size_t)r * OUT_CH] = acc[r];
}

// ---------------------------------------------------------------------------
extern "C" void kernel_launch(void* const* d_in, const int* in_sizes, int n_in,
                              void* d_out, int out_size, void* d_ws, size_t ws_size,
                              hipStream_t stream)
{
    const float* z  = (const float*)d_in[0];
    const float* wx = (const float*)d_in[1];
    const float* bx = (const float*)d_in[2];
    const float* wy = (const float*)d_in[3];
    const float* by = (const float*)d_in[4];
    float* out = (float*)d_out;
    float* zm  = (float*)d_ws;                 // 64*512 floats = 128 KB scratch

    simcc_mean_kernel<<<BATCH * IN_CH, 256, 0, stream>>>(z, zm);
    simcc_gemm_kernel<<<672, 128, 0, stream>>>(zm, wx, bx, wy, by, out);
}